// SVDDLayer_74801150427722
// MI455X (gfx1250) — compile-verified
//
#include <hip/hip_runtime.h>
#include <math.h>

typedef float v2f __attribute__((ext_vector_type(2)));
typedef float v8f __attribute__((ext_vector_type(8)));

#define QP_N    10
#define FEAT    256
#define EPS_F   1e-6f
#define SIGMA_F 0.1f
#define N_ITERS 30

__host__ __device__ __forceinline__ constexpr int tri(int i) { return i * (i + 1) / 2; }

// ---------------------------------------------------------------------------
// Kernel 1: per-batch Gram matrix K = X X^T (+ EPS on diag) via fp32 WMMA.
// One wave per batch. A and B operands share the same register pair (X Xᵀ).
// Lanes owning pad rows (10-15) load a clamped valid row instead of being
// EXEC-masked: D rows/cols 10-15 become garbage but are never stored, and the
// loop stays branch-free (no saveexec per load). Loads are batched 8-deep so
// WMMAs overlap outstanding global_load_b64s instead of waiting one-by-one.
// ---------------------------------------------------------------------------
__global__ void gram_wmma_kernel(const float* __restrict__ X,
                                 float* __restrict__ Kout, int B) {
    const int lane = threadIdx.x & 31;
    const int wave = threadIdx.x >> 5;
    const int b = blockIdx.x * (blockDim.x >> 5) + wave;
    if (b >= B) return;                      // wave-uniform: EXEC stays all-1s

    const int row   = lane & 15;             // M (and N) index owned by lane
    const int khalf = (lane >> 4) << 1;      // lanes 16-31 hold K+2,K+3
    const int rowc  = (row < QP_N) ? row : (QP_N - 1);   // clamp pad rows
    const float* xb = X + (size_t)b * QP_N * FEAT + (size_t)rowc * FEAT + khalf;

    v8f acc = {};
#pragma unroll 1
    for (int k = 0; k < FEAT; k += 32) {     // 8 WMMA steps per trip
        v2f ld[8];
#pragma unroll
        for (int u = 0; u < 8; ++u)
            ld[u] = *(const v2f*)(xb + k + 4 * u);
#pragma unroll
        for (int u = 0; u < 8; ++u) {
            // A 16x4 fp32 layout: V0 = {K | K+2}, V1 = {K+1 | K+3};
            // B 4x16 layout is the mirror -> same registers for X·Xᵀ.
            acc = __builtin_amdgcn_wmma_f32_16x16x4_f32(
                /*neg_a=*/false, ld[u], /*neg_b=*/false, ld[u],
                /*c_mod=*/(short)0, acc, /*reuse_a=*/false, /*reuse_b=*/false);
        }
    }

    // C/D layout: acc[r] holds element (M = r + 8*(lane>=16), N = lane&15)
    const int n = lane & 15;
    const int mbase = (lane >> 4) << 3;
    if (n < QP_N) {
#pragma unroll
        for (int r = 0; r < 8; ++r) {
            const int m = mbase + r;
            if (m < QP_N) {
                float v = acc[r] + ((m == n) ? EPS_F : 0.0f);
                Kout[(size_t)b * 100 + m * QP_N + n] = v;
            }
        }
    }
}

// ---------------------------------------------------------------------------
// Kernel 2: one thread per batch. 30 primal-dual IP iterations.
// KKT system [[M,1],[1ᵀ,0]] solved by block elimination with Cholesky of the
// SPD matrix M = Q + diag(lam/a). All n=10 loops fully unrolled -> registers.
// ---------------------------------------------------------------------------
__global__ void qp_solve_kernel(const float* __restrict__ Kmat,
                                float* __restrict__ alph, int B) {
    const int b = blockIdx.x * blockDim.x + threadIdx.x;
    if (b >= B) return;
    const float* kb = Kmat + (size_t)b * 100;

    // Q = 2*(K + EPS I), lower triangle only (symmetric)
    float Qt[55];
#pragma unroll
    for (int i = 0; i < QP_N; ++i)
#pragma unroll
        for (int j = 0; j <= i; ++j)
            Qt[tri(i) + j] = 2.0f * kb[i * QP_N + j];

    float p[QP_N];
#pragma unroll
    for (int i = 0; i < QP_N; ++i) p[i] = -0.5f * Qt[tri(i) + i]; // -diag(K)

    float a[QP_N], lam[QP_N];
#pragma unroll
    for (int i = 0; i < QP_N; ++i) { a[i] = 1.0f / QP_N; lam[i] = 1.0f; }
    float nu = 0.0f;

#pragma unroll 1
    for (int it = 0; it < N_ITERS; ++it) {
        float mu = 0.0f;
#pragma unroll
        for (int i = 0; i < QP_N; ++i) mu += a[i] * lam[i];
        mu *= (1.0f / QP_N);

        float rprim = -1.0f;
#pragma unroll
        for (int i = 0; i < QP_N; ++i) rprim += a[i];

        // rhs_top = -(Q a + p + nu - lam) + SIGMA*mu/a - lam
        float r1[QP_N];
#pragma unroll
        for (int i = 0; i < QP_N; ++i) {
            float s = 0.0f;
#pragma unroll
            for (int j = 0; j < QP_N; ++j) {
                const float q = (i >= j) ? Qt[tri(i) + j] : Qt[tri(j) + i];
                s = fmaf(q, a[j], s);
            }
            r1[i] = -(s + p[i] + nu - lam[i]) + SIGMA_F * mu / a[i] - lam[i];
        }

        // M = Q + diag(lam/a); in-place Cholesky M = L Lᵀ
        float Lm[55];
#pragma unroll
        for (int k = 0; k < 55; ++k) Lm[k] = Qt[k];
#pragma unroll
        for (int i = 0; i < QP_N; ++i) Lm[tri(i) + i] += lam[i] / a[i];

        float invd[QP_N];
#pragma unroll
        for (int j = 0; j < QP_N; ++j) {
            float s = Lm[tri(j) + j];
#pragma unroll
            for (int k = 0; k < j; ++k) s = fmaf(-Lm[tri(j) + k], Lm[tri(j) + k], s);
            const float d = sqrtf(s);
            Lm[tri(j) + j] = d;
            invd[j] = 1.0f / d;
#pragma unroll
            for (int i = j + 1; i < QP_N; ++i) {
                float t = Lm[tri(i) + j];
#pragma unroll
                for (int k = 0; k < j; ++k) t = fmaf(-Lm[tri(i) + k], Lm[tri(j) + k], t);
                Lm[tri(i) + j] = t * invd[j];
            }
        }

        // Solve M x = r1 and M y = 1 (forward then backward substitution)
        float x[QP_N], y[QP_N];
#pragma unroll
        for (int i = 0; i < QP_N; ++i) {
            float sx = r1[i], sy = 1.0f;
#pragma unroll
            for (int k = 0; k < i; ++k) {
                sx = fmaf(-Lm[tri(i) + k], x[k], sx);
                sy = fmaf(-Lm[tri(i) + k], y[k], sy);
            }
            x[i] = sx * invd[i];
            y[i] = sy * invd[i];
        }
#pragma unroll
        for (int i = QP_N - 1; i >= 0; --i) {
            float sx = x[i], sy = y[i];
#pragma unroll
            for (int k = i + 1; k < QP_N; ++k) {
                sx = fmaf(-Lm[tri(k) + i], x[k], sx);
                sy = fmaf(-Lm[tri(k) + i], y[k], sy);
            }
            x[i] = sx * invd[i];
            y[i] = sy * invd[i];
        }

        float sumx = 0.0f, sumy = 0.0f;
#pragma unroll
        for (int i = 0; i < QP_N; ++i) { sumx += x[i]; sumy += y[i]; }
        // dnu = (1ᵀ M⁻¹ rhs_top − rhs_last) / (1ᵀ M⁻¹ 1),  rhs_last = −r_prim
        const float dnu = (sumx + rprim) / sumy;

        float da[QP_N], dlam[QP_N];
#pragma unroll
        for (int i = 0; i < QP_N; ++i) da[i] = fmaf(-dnu, y[i], x[i]);
#pragma unroll
        for (int i = 0; i < QP_N; ++i)
            dlam[i] = (SIGMA_F * mu - a[i] * lam[i] - lam[i] * da[i]) / a[i];

        float ta = 3.0e38f, tl = 3.0e38f;
#pragma unroll
        for (int i = 0; i < QP_N; ++i) {
            if (da[i]   < 0.0f) ta = fminf(ta, -a[i]   / da[i]);
            if (dlam[i] < 0.0f) tl = fminf(tl, -lam[i] / dlam[i]);
        }
        const float t = fminf(1.0f, 0.99f * fminf(ta, tl));

#pragma unroll
        for (int i = 0; i < QP_N; ++i) {
            a[i]   = fmaf(t, da[i],   a[i]);
            lam[i] = fmaf(t, dlam[i], lam[i]);
        }
        nu = fmaf(t, dnu, nu);
    }

#pragma unroll
    for (int i = 0; i < QP_N; ++i) alph[(size_t)b * QP_N + i] = a[i];
}

// ---------------------------------------------------------------------------
// Kernel 3: centers[b,d] = sum_s alpha[b,s] * X[b,s,d].  Block per batch,
// thread per feature -> fully coalesced; X re-read hits L2 (84MB < 192MB).
// ---------------------------------------------------------------------------
__global__ void centers_kernel(const float* __restrict__ X,
                               const float* __restrict__ alph,
                               float* __restrict__ out) {
    const int b = blockIdx.x;
    const int d = threadIdx.x;
    const float* xb = X + (size_t)b * QP_N * FEAT + d;
    const float* ab = alph + (size_t)b * QP_N;
    float s = 0.0f;
#pragma unroll
    for (int j = 0; j < QP_N; ++j) s = fmaf(ab[j], xb[(size_t)j * FEAT], s);
    out[(size_t)b * FEAT + d] = s;
}

// ---------------------------------------------------------------------------
extern "C" void kernel_launch(void* const* d_in, const int* in_sizes, int n_in,
                              void* d_out, int out_size, void* d_ws, size_t ws_size,
                              hipStream_t stream) {
    const float* X = (const float*)d_in[0];
    const int B = in_sizes[0] / (QP_N * FEAT);

    float* Kws = (float*)d_ws;                  // B*100 floats
    float* Aws = Kws + (size_t)B * 100;         // B*10 floats
    float* out = (float*)d_out;

    {   // Gram via WMMA: 8 waves (= 8 batches) per 256-thread block
        const int wpb = 8;
        gram_wmma_kernel<<<dim3((B + wpb - 1) / wpb), dim3(32 * wpb), 0, stream>>>(X, Kws, B);
    }
    {   // Interior-point QP: one thread per batch
        qp_solve_kernel<<<dim3((B + 255) / 256), dim3(256), 0, stream>>>(Kws, Aws, B);
    }
    {   // Weighted centers
        centers_kernel<<<dim3(B), dim3(FEAT), 0, stream>>>(X, Aws, out);
    }
}